// ContextMultiheadAttention_23063974379831
// MI455X (gfx1250) — compile-verified
//
#include <hip/hip_runtime.h>
#include <cstdint>

typedef __attribute__((ext_vector_type(2))) float v2f;
typedef __attribute__((ext_vector_type(8))) float v8f;

#define N_PTS   256
#define HEADS   8
#define DMODEL  256
#define DHEAD   32
#define LEVELS  16
#define TSIZE   4096
#define TMASK   4095
// log2(20)/15
#define LOG2B   0.2881285397f

// ---------------------------------------------------------------------------
// WMMA f32 GEMM: C(MxN) = A(MxK) * B(KxN), one 16x16 tile per wave32,
// k-step 4 via V_WMMA_F32_16X16X4_F32.
//   mode 0: plain store to C
//   mode 1: QW epilogue  -> qb with faithful-reshape permutation
//   mode 2: KV epilogue  -> kb (cols<256) / vb (cols>=256), same permutation
// Fragment layout (wave32):
//   A 16x4 : lane(lo=lane&15,hi=lane>>4): a.x=A[lo][k+2hi], a.y=A[lo][k+2hi+1]
//   B 4x16 : b.x=B[k+2hi][n], b.y=B[k+2hi+1][n]
//   C 16x16: c[r]=C[r+8hi][lo]
// ---------------------------------------------------------------------------
__global__ __launch_bounds__(256)
void gemm_wmma_f32(const float* __restrict__ A, const float* __restrict__ B,
                   float* __restrict__ C,
                   float* __restrict__ qb, float* __restrict__ kb,
                   float* __restrict__ vb,
                   int M, int N, int K, int mode)
{
    const int gwave  = (blockIdx.x * 256 + threadIdx.x) >> 5;
    const int lane   = threadIdx.x & 31;
    const int tilesN = N >> 4;
    const int tiles  = (M >> 4) * tilesN;
    if (gwave >= tiles) return;
    const int tm = gwave / tilesN;
    const int tn = gwave % tilesN;
    const int lo = lane & 15;
    const int hi = lane >> 4;

    v8f acc = {0.f, 0.f, 0.f, 0.f, 0.f, 0.f, 0.f, 0.f};
    const float* Arow = A + (size_t)(tm * 16 + lo) * K;
    const int    cb   = tn * 16 + lo;

    for (int k0 = 0; k0 < K; k0 += 4) {
        v2f a, b;
        a.x = Arow[k0 + 2 * hi];
        a.y = Arow[k0 + 2 * hi + 1];
        b.x = B[(size_t)(k0 + 2 * hi) * N + cb];
        b.y = B[(size_t)(k0 + 2 * hi + 1) * N + cb];
        acc = __builtin_amdgcn_wmma_f32_16x16x4_f32(
                  false, a, false, b, (short)0, acc, false, false);
    }

#pragma unroll
    for (int r8 = 0; r8 < 8; ++r8) {
        const int r   = tm * 16 + 8 * hi + r8;
        const int col = cb;
        const float val = acc[r8];
        if (mode == 0) {
            C[(size_t)r * N + col] = val;
        } else {
            float* dst;
            int c2 = col;
            if (mode == 1) {
                dst = qb;
            } else {
                if (c2 < DMODEL) dst = kb;
                else { dst = vb; c2 -= DMODEL; }
            }
            // faithful flat reshape (N,D)->(H,N,d):
            // dst[h,n,dd] = SRC[h*32 + n/8, (n%8)*32 + dd]
            const int h  = r >> 5;
            const int n  = ((r & 31) << 3) + (c2 >> 5);
            const int dd = c2 & 31;
            dst[((h * N_PTS + n) << 5) + dd] = val;
        }
    }
}

// ---------------------------------------------------------------------------
// Fused hash-encode + bias attention + online softmax + PV.
// Grid: one block per query n. Block: 256 threads = 8 wave32, wave == head.
// Loop over 8 key tiles of 32:
//   Phase A (all waves): 512 (m,level) tasks -> trilinear hash gather of the
//     24 tables (shared hash/weights) into LDS enc tiles QE/KE/VE [8][32][33].
//   Phase B (wave h, lane = m): att = q·k + q·q_enc + k·k_enc, flash-softmax
//     running (max,sum), yacc[dd=lane] += p * (v + v_enc) via lane broadcast.
// ---------------------------------------------------------------------------
__global__ __launch_bounds__(256)
void attn_hash_kernel(const float* __restrict__ xyz,
                      const float* __restrict__ tables,
                      const float* __restrict__ qb,
                      const float* __restrict__ kb,
                      const float* __restrict__ vb,
                      float* __restrict__ yb)
{
    extern __shared__ float smem[];
    float* QE = smem;                 // 8*32*33 = 8448
    float* KE = smem + 8448;
    float* VE = smem + 16896;
    float* QS = smem + 25344;         // 256 staged q values
    const int n    = blockIdx.x;
    const int tid  = threadIdx.x;
    const int h    = tid >> 5;
    const int lane = tid & 31;
    const float2* tab2 = (const float2*)tables;   // [(e*16+l)*4096 + idx]

    QS[tid] = qb[((h * N_PTS + n) << 5) + lane];

    const float xn = xyz[n * 3 + 0];
    const float yn = xyz[n * 3 + 1];
    const float zn = xyz[n * 3 + 2];

    float run_max = -INFINITY;
    float run_sum = 0.f;
    float yacc    = 0.f;

    for (int tile = 0; tile < 8; ++tile) {
        __syncthreads();   // previous tile's phase B done before overwrite
        // ---------------- Phase A: build enc tiles in LDS ----------------
        for (int t = tid; t < 512; t += 256) {
            const int m  = t & 31;
            const int l  = t >> 5;
            const int mg = tile * 32 + m;
            const float res = exp2f((float)l * LOG2B);
            const float sx = (xn - xyz[mg * 3 + 0]) * res;
            const float sy = (yn - xyz[mg * 3 + 1]) * res;
            const float sz = (zn - xyz[mg * 3 + 2]) * res;
            const float fx = floorf(sx), fy = floorf(sy), fz = floorf(sz);
            const float wx = sx - fx, wy = sy - fy, wz = sz - fz;
            const uint32_t ix = (uint32_t)(int32_t)fx;
            const uint32_t iy = (uint32_t)(int32_t)fy;
            const uint32_t iz = (uint32_t)(int32_t)fz;

            float accx[24], accy[24];
#pragma unroll
            for (int e = 0; e < 24; ++e) { accx[e] = 0.f; accy[e] = 0.f; }

            for (int c = 0; c < 8; ++c) {
                const uint32_t cx = ix + (uint32_t)(c & 1);
                const uint32_t cy = iy + (uint32_t)((c >> 1) & 1);
                const uint32_t cz = iz + (uint32_t)((c >> 2) & 1);
                const uint32_t hsh = cx ^ (cy * 2654435761u) ^ (cz * 805459861u);
                const int idx = (int)(hsh & TMASK);
                const float wc = ((c & 1)        ? wx : 1.f - wx)
                               * (((c >> 1) & 1) ? wy : 1.f - wy)
                               * (((c >> 2) & 1) ? wz : 1.f - wz);
                const float2* base = tab2 + (size_t)l * TSIZE + idx;
#pragma unroll
                for (int e = 0; e < 24; ++e) {
                    const float2 tv = base[(size_t)e * (LEVELS * TSIZE)];
                    accx[e] += wc * tv.x;
                    accy[e] += wc * tv.y;
                }
            }
            const int d0 = 2 * l;
#pragma unroll
            for (int hh = 0; hh < 8; ++hh) {
                const int row = (hh * 32 + m) * 33;
                QE[row + d0]     = accx[hh];      QE[row + d0 + 1] = accy[hh];
                KE[row + d0]     = accx[8 + hh];  KE[row + d0 + 1] = accy[8 + hh];
                VE[row + d0]     = accx[16 + hh]; VE[row + d0 + 1] = accy[16 + hh];
            }
        }
        __syncthreads();
        // ---------------- Phase B: attention over this key tile ----------
        const int mg = tile * 32 + lane;
        const float* krow = kb + ((size_t)(h * N_PTS + mg) << 5);
        const float* qrow = QS + h * DHEAD;
        const float* qe   = QE + (h * 32 + lane) * 33;
        const float* ke   = KE + (h * 32 + lane) * 33;
        float att = 0.f;
#pragma unroll 8
        for (int dd = 0; dd < 32; ++dd) {
            const float qv = qrow[dd];
            const float kv = krow[dd];
            att += qv * kv + qv * qe[dd] + kv * ke[dd];
        }
        float tmax = att;
#pragma unroll
        for (int i = 16; i > 0; i >>= 1) tmax = fmaxf(tmax, __shfl_xor(tmax, i));
        const float nmax  = fmaxf(run_max, tmax);
        const float scale = __expf(run_max - nmax);
        const float p     = __expf(att - nmax);
        float tsum = p;
#pragma unroll
        for (int i = 16; i > 0; i >>= 1) tsum += __shfl_xor(tsum, i);
        run_sum = run_sum * scale + tsum;
        yacc   *= scale;
        const float* vbase  = vb + ((size_t)(h * N_PTS + tile * 32) << 5) + lane;
        const float* vebase = VE + h * 32 * 33 + lane;
#pragma unroll 4
        for (int mp = 0; mp < 32; ++mp) {
            const float pb = __shfl(p, mp);
            yacc += pb * (vbase[mp * DHEAD] + vebase[mp * 33]);
        }
        run_max = nmax;
    }
    // y.swapaxes(0,1).reshape(N,D): row n, col h*32+dd
    yb[n * DMODEL + h * DHEAD + lane] = yacc / run_sum;
}

// ---------------------------------------------------------------------------
extern "C" void kernel_launch(void* const* d_in, const int* in_sizes, int n_in,
                              void* d_out, int out_size, void* d_ws, size_t ws_size,
                              hipStream_t stream) {
    const float* xyz    = (const float*)d_in[0];   // (256,3)
    const float* feat   = (const float*)d_in[1];   // (256,256)
    const float* tables = (const float*)d_in[2];   // (3,8,16,4096,2)
    const float* Wq     = (const float*)d_in[3];   // (256,256)
    const float* Wkv    = (const float*)d_in[4];   // (256,512)
    const float* Wproj  = (const float*)d_in[5];   // (256,256)
    float* out = (float*)d_out;                    // (256,256)

    // workspace: q,k,v in (H,N,32) layout + y in (N,256); 1 MB total
    float* qb = (float*)d_ws;
    float* kb = qb + HEADS * N_PTS * DHEAD;        // +65536
    float* vb = kb + HEADS * N_PTS * DHEAD;        // +131072
    float* yb = vb + HEADS * N_PTS * DHEAD;        // +196608
    (void)in_sizes; (void)n_in; (void)out_size; (void)ws_size;

    // feat @ Wq : 256x256x256 -> qb (permuted); 256 tiles / 8 waves = 32 blocks
    gemm_wmma_f32<<<32, 256, 0, stream>>>(feat, Wq, nullptr, qb, kb, vb,
                                          DMODEL, DMODEL, DMODEL, 1);
    // feat @ Wkv : 256x512x256 -> kb/vb (permuted); 512 tiles -> 64 blocks
    gemm_wmma_f32<<<64, 256, 0, stream>>>(feat, Wkv, nullptr, qb, kb, vb,
                                          DMODEL, 2 * DMODEL, DMODEL, 2);
    // fused hash-encode + attention: one block per query point
    attn_hash_kernel<<<N_PTS, 256, 102400, stream>>>(xyz, tables, qb, kb, vb, yb);
    // y @ Wproj : 256x256x256 -> out
    gemm_wmma_f32<<<32, 256, 0, stream>>>(yb, Wproj, out, nullptr, nullptr, nullptr,
                                          DMODEL, DMODEL, DMODEL, 0);
}